// Embedding_27685359190360
// MI455X (gfx1250) — compile-verified
//
#include <hip/hip_runtime.h>

// LoRA embedding: out[t, :] = A[idx[t], :] (1x16) @ B (16x1024), fp32.
// Memory-bound on the 256 MB output; compute done with chained
// V_WMMA_F32_16X16X4_F32 (K=16 as 4 x K=4 steps), wave32.

typedef __attribute__((ext_vector_type(2))) float v2f;
typedef __attribute__((ext_vector_type(8))) float v8f;

#define RANK 16
#define DIM 1024
#define NTOK 65536               // 32 * 2048 tokens
#define TILES (NTOK / 16)        // 4096 token tiles of 16
#define NCHUNKS (DIM / 16)       // 64 column chunks of 16
#define WAVES_PER_BLOCK 8
#define TILES_PER_WAVE 8         // 64 groups * 8 waves * 8 tiles = 4096

__global__ __launch_bounds__(256)
void lora_embed_wmma_f32(const int* __restrict__ idx,
                         const float* __restrict__ A,
                         const float* __restrict__ B,
                         float* __restrict__ out) {
    const int lane   = threadIdx.x & 31;
    const int wave   = threadIdx.x >> 5;
    const int nchunk = blockIdx.x;          // [0, 64): which 16-col slab
    const int nlo    = lane & 15;           // N (and M-row-select for A gather)
    const int khalf  = lane >> 4;           // 0: K pair {0,1}, 1: K pair {2,3}
    const int col    = nchunk * 16 + nlo;

    // ---- B fragments: 4 x (4x16) K-slices, loaded once, reused 8 tiles ----
    // Layout per ISA: VGPR0 = K0 (lanes 0-15) / K2 (lanes 16-31), VGPR1 = K1/K3.
    v2f bfrag[4];
#pragma unroll
    for (int f = 0; f < 4; ++f) {
        const int k0 = 4 * f + 2 * khalf;
        bfrag[f].x = B[(size_t)k0 * DIM + col];
        bfrag[f].y = B[(size_t)(k0 + 1) * DIM + col];
    }

    const int tile0 = (blockIdx.y * WAVES_PER_BLOCK + wave) * TILES_PER_WAVE;

#pragma unroll 1
    for (int t = 0; t < TILES_PER_WAVE; ++t) {
        const int token_base = (tile0 + t) * 16;

        // Gathered A row for this lane's M (= nlo); 16 fp32 = 64 B aligned.
        const int row = idx[token_base + nlo];
        const float* __restrict__ arow = A + (size_t)row * RANK;

        // A fragments: lane M = nlo; lanes 0-15 hold K={4f,4f+1},
        // lanes 16-31 hold K={4f+2,4f+3}  ->  one b64 load each.
        v2f afrag[4];
#pragma unroll
        for (int f = 0; f < 4; ++f) {
            const int k0 = 4 * f + 2 * khalf;
            afrag[f].x = arow[k0];
            afrag[f].y = arow[k0 + 1];
        }

        // D = sum_f A_f (16x4) * B_f (4x16), fp32 accumulate.
        v8f c = {};
#pragma unroll
        for (int f = 0; f < 4; ++f) {
            c = __builtin_amdgcn_wmma_f32_16x16x4_f32(
                    /*neg_a=*/false, afrag[f],
                    /*neg_b=*/false, bfrag[f],
                    /*c_mod=*/(short)0, c,
                    /*reuse_a=*/false, /*reuse_b=*/false);
        }

        // C/D layout: VGPR v -> M = v + 8*khalf, N = nlo.
        float* __restrict__ orow =
            out + (size_t)(token_base + 8 * khalf) * DIM + col;
#pragma unroll
        for (int v = 0; v < 8; ++v) {
            orow[(size_t)v * DIM] = c[v];
        }
    }
}

extern "C" void kernel_launch(void* const* d_in, const int* in_sizes, int n_in,
                              void* d_out, int out_size, void* d_ws, size_t ws_size,
                              hipStream_t stream) {
    const int*   idx = (const int*)d_in[0];     // [32, 2048] int32
    const float* A   = (const float*)d_in[1];   // [128000, 16] fp32
    const float* B   = (const float*)d_in[2];   // [16, 1024] fp32
    float*       out = (float*)d_out;           // [32, 2048, 1024] fp32

    dim3 grid(NCHUNKS, TILES / (WAVES_PER_BLOCK * TILES_PER_WAVE));  // (64, 64)
    lora_embed_wmma_f32<<<grid, 256, 0, stream>>>(idx, A, B, out);
}